// Pool_76519137345590
// MI455X (gfx1250) — compile-verified
//
#include <hip/hip_runtime.h>

#define B_  16
#define N_  4096
#define D_  128
#define M_  1024
#define R2_ (0.2f * 0.2f)

typedef float v2f __attribute__((ext_vector_type(2)));
typedef float v8f __attribute__((ext_vector_type(8)));
typedef unsigned long long u64;
typedef unsigned int u32;

// ---------------------------------------------------------------------------
// Kernel 1: farthest-point sampling, one workgroup per graph.
// 256 threads, 16 points/thread held entirely in VGPRs. Per round:
//   local dist update + local argmax (registers) ->
//   wave32 shfl_xor max-reduce -> 8 keys in LDS -> thread0 picks winner,
//   broadcasts winner coords + writes sampled outputs.
// Key packs (dist_bits << 32) | (N-1-idx) so max-key == (max dist, min idx),
// matching jnp.argmax tie-breaking.
// ---------------------------------------------------------------------------
__global__ __launch_bounds__(256) void fps_kernel(
    const float* __restrict__ pos,
    int* __restrict__ idx_ws, float4* __restrict__ pos4_ws,
    float* __restrict__ out_pos, float* __restrict__ out_batch)
{
    __shared__ float sx_[N_], sy_[N_], sz_[N_];
    __shared__ u64   wbest[8];
    __shared__ float bx, by, bz;

    const int b = blockIdx.x;
    const int tid = threadIdx.x;
    const float* pg = pos + (size_t)b * N_ * 3;

    for (int i = tid; i < N_; i += 256) {
        sx_[i] = pg[i * 3 + 0];
        sy_[i] = pg[i * 3 + 1];
        sz_[i] = pg[i * 3 + 2];
    }
    __syncthreads();

    float px[16], py[16], pz[16], dist[16];
#pragma unroll
    for (int p = 0; p < 16; ++p) {
        const int i = p * 256 + tid;
        px[p] = sx_[i]; py[p] = sy_[i]; pz[p] = sz_[i];
        dist[p] = 3.4e38f;
    }

    if (tid == 0) {
        const float nx = sx_[0], ny = sy_[0], nz = sz_[0];
        bx = nx; by = ny; bz = nz;
        const int r = b * M_;
        idx_ws[r] = 0;
        pos4_ws[r] = make_float4(nx, ny, nz, nx * nx + ny * ny + nz * nz);
        out_pos[(size_t)r * 3 + 0] = nx;
        out_pos[(size_t)r * 3 + 1] = ny;
        out_pos[(size_t)r * 3 + 2] = nz;
        out_batch[r] = (float)b;
    }
    __syncthreads();

    const int lane = tid & 31;
    const int wid  = tid >> 5;

    for (int m = 1; m < M_; ++m) {
        const float cx = bx, cy = by, cz = bz;   // coords of selection m-1

        float bd = -1.0f; int bi = 0;
#pragma unroll
        for (int p = 0; p < 16; ++p) {
            const float dx = px[p] - cx, dy = py[p] - cy, dz = pz[p] - cz;
            const float d  = dx * dx + dy * dy + dz * dz;
            const float nd = fminf(dist[p], d);
            dist[p] = nd;
            if (nd > bd) { bd = nd; bi = p * 256 + tid; }
        }

        u64 key = ((u64)__float_as_uint(bd) << 32) | (u32)(N_ - 1 - bi);
#pragma unroll
        for (int off = 16; off > 0; off >>= 1) {
            const u64 o = __shfl_xor(key, off, 32);
            key = (o > key) ? o : key;
        }
        if (lane == 0) wbest[wid] = key;
        __syncthreads();

        if (tid == 0) {
            u64 k = wbest[0];
#pragma unroll
            for (int w = 1; w < 8; ++w) k = (wbest[w] > k) ? wbest[w] : k;
            const int sel = N_ - 1 - (int)(u32)(k & 0xffffffffu);
            const float nx = sx_[sel], ny = sy_[sel], nz = sz_[sel];
            bx = nx; by = ny; bz = nz;
            const int r = b * M_ + m;
            idx_ws[r] = sel;
            pos4_ws[r] = make_float4(nx, ny, nz, nx * nx + ny * ny + nz * nz);
            out_pos[(size_t)r * 3 + 0] = nx;
            out_pos[(size_t)r * 3 + 1] = ny;
            out_pos[(size_t)r * 3 + 2] = nz;
            out_batch[r] = (float)b;
        }
        __syncthreads();
    }
}

// ---------------------------------------------------------------------------
// Kernel 2: gather sampled features. 32 lanes per row, float4 streaming.
// ---------------------------------------------------------------------------
__global__ __launch_bounds__(256) void gather_kernel(
    const float* __restrict__ x,
    const int* __restrict__ idx_ws,
    float* __restrict__ out_x)
{
    const int t = blockIdx.x * 256 + threadIdx.x;
    const int r = t >> 5;               // sampled row index [0, B*M)
    const int c = (t & 31) << 2;        // float offset within row
    const int b = r >> 10;              // r / M_
    const int src = idx_ws[r];
    const float4 v = *(const float4*)(x + ((size_t)b * N_ + src) * D_ + c);
    *(float4*)(out_x + (size_t)r * D_ + c) = v;
}

// ---------------------------------------------------------------------------
// Kernel 3: radius adjacency via WMMA Gram matrix.
// One wave per 16x16 tile: d2 = ni + nj - 2*<pi,pj>,
// <pi,pj> from v_wmma_f32_16x16x4_f32 (K padded 3 -> 4 with zeros).
// A layout per ISA: lanes 0-15 carry K=0,1 ; lanes 16-31 carry K=2,3.
// Branch-free operand construction keeps EXEC all-ones at the WMMA.
// ---------------------------------------------------------------------------
__global__ __launch_bounds__(256) void adj_kernel(
    const float4* __restrict__ pos4,
    float* __restrict__ adj)
{
    const int tid  = threadIdx.x;
    const int wave = tid >> 5;
    const int lane = tid & 31;
    const int tile = blockIdx.x * 8 + wave;     // B * 64 * 64 tiles
    const int b  = tile >> 12;
    const int ti = (tile >> 6) & 63;
    const int tj = tile & 63;
    const int i0 = ti << 4;
    const int j0 = tj << 4;
    const int l    = lane & 15;
    const int half = lane >> 4;

    const float4* pg = pos4 + b * M_;
    const float4 pi = pg[i0 + l];
    const float4 pj = pg[j0 + l];

    v2f a, bv;
    a.x  = half ? pi.z : pi.x;
    a.y  = half ? 0.0f : pi.y;
    bv.x = half ? pj.z : pj.x;
    bv.y = half ? 0.0f : pj.y;

    v8f c = {};
    c = __builtin_amdgcn_wmma_f32_16x16x4_f32(
            /*neg_a=*/false, a, /*neg_b=*/false, bv,
            /*c_mod=*/(short)0, c, /*reuse_a=*/false, /*reuse_b=*/false);

    const float nj  = pj.w;            // |p_col|^2, col n = j0 + l (C layout)
    const int   ncol = j0 + l;
    float* arow = adj + (size_t)b * M_ * M_;

#pragma unroll
    for (int r = 0; r < 8; ++r) {
        const int mrow = i0 + half * 8 + r;        // C layout: row M per VGPR
        const float ni = pg[mrow].w;
        const float d2 = ni + nj - 2.0f * c[r];
        const float v  = (d2 <= R2_ && mrow != ncol) ? 1.0f : 0.0f;
        arow[(size_t)mrow * M_ + ncol] = v;
    }
}

// ---------------------------------------------------------------------------
extern "C" void kernel_launch(void* const* d_in, const int* in_sizes, int n_in,
                              void* d_out, int out_size, void* d_ws, size_t ws_size,
                              hipStream_t stream)
{
    const float* x   = (const float*)d_in[0];   // [B*N, D]
    const float* pos = (const float*)d_in[1];   // [B*N, 3]
    // d_in[2] (batch) not needed: equal-size sorted graphs.

    float* out       = (float*)d_out;
    float* out_x     = out;                                   // B*M*D
    float* out_pos   = out_x + (size_t)B_ * M_ * D_;          // B*M*3
    float* out_batch = out_pos + (size_t)B_ * M_ * 3;         // B*M
    float* out_adj   = out_batch + (size_t)B_ * M_;           // B*M*M

    int*    idx_ws  = (int*)d_ws;                                   // B*M ints
    float4* pos4_ws = (float4*)((char*)d_ws + (size_t)B_ * M_ * 4); // B*M float4

    fps_kernel<<<B_, 256, 0, stream>>>(pos, idx_ws, pos4_ws, out_pos, out_batch);
    gather_kernel<<<(B_ * M_ * 32) / 256, 256, 0, stream>>>(x, idx_ws, out_x);
    adj_kernel<<<(B_ * 64 * 64) / 8, 256, 0, stream>>>(pos4_ws, out_adj);
}